// LightGCN_89094801588748
// MI455X (gfx1250) — compile-verified
//
#include <hip/hip_runtime.h>

#define USER_COUNT 200000
#define ITEM_COUNT 100000
#define NROWS (USER_COUNT + ITEM_COUNT)
#define DIM 64
#define NUM_LAYER 3

typedef float v2f __attribute__((ext_vector_type(2)));
typedef float v8f __attribute__((ext_vector_type(8)));

// -------- init: acc = cur = concat(user_emb, item_emb); nxt = 0 --------
__global__ void lgcn_init(const float4* __restrict__ ue, const float4* __restrict__ ie,
                          float4* __restrict__ acc, float4* __restrict__ cur,
                          float4* __restrict__ nxt) {
    const long total = (long)NROWS * (DIM / 4);
    const long utot  = (long)USER_COUNT * (DIM / 4);
    for (long i = (long)blockIdx.x * blockDim.x + threadIdx.x; i < total;
         i += (long)gridDim.x * blockDim.x) {
        float4 v = (i < utot) ? ue[i] : ie[i - utot];
        acc[i] = v;
        cur[i] = v;
        nxt[i] = make_float4(0.f, 0.f, 0.f, 0.f);
    }
}

// -------- SpMM: one wave32 per nonzero; lane handles 2 dims (float2 gather + 2 atomics)
__global__ void lgcn_spmm(const int* __restrict__ row, const int* __restrict__ col,
                          const float* __restrict__ val, const float* __restrict__ x,
                          float* __restrict__ y, int nnz) {
    int gid  = blockIdx.x * blockDim.x + threadIdx.x;
    int e    = gid >> 5;
    int lane = gid & 31;
    if (e >= nnz) return;
    int   r = row[e];
    int   c = col[e];
    float v = val[e];
    const float2* xr = (const float2*)(x + (size_t)c * DIM);
    float2 m = xr[lane];                       // coalesced 256B row gather per wave
    float* yr = y + (size_t)r * DIM;
    atomicAdd(yr + 2 * lane,     v * m.x);     // global_atomic_add_f32, lands in L2
    atomicAdd(yr + 2 * lane + 1, v * m.y);
}

// -------- combine: acc += nxt; zero the buffer that becomes next layer's target --------
__global__ void lgcn_combine(float4* __restrict__ acc, const float4* __restrict__ nxt,
                             float4* __restrict__ zbuf) {
    const long total = (long)NROWS * (DIM / 4);
    for (long i = (long)blockIdx.x * blockDim.x + threadIdx.x; i < total;
         i += (long)gridDim.x * blockDim.x) {
        float4 a = acc[i];
        float4 n = nxt[i];
        a.x += n.x; a.y += n.y; a.z += n.z; a.w += n.w;
        acc[i]  = a;
        zbuf[i] = make_float4(0.f, 0.f, 0.f, 0.f);
    }
}

// -------- scoring: one wave32 per 16 batch rows; diag(U*I^T), diag(U*J^T) via
// V_WMMA_F32_16X16X4_F32 chained over K=64. fp32 end-to-end, matching reference.
__global__ void lgcn_score(const float* __restrict__ acc, const int* __restrict__ user,
                           const int* __restrict__ pos, const int* __restrict__ neg,
                           float* __restrict__ out, int B) {
    int lane = threadIdx.x;        // blockDim.x == 32, exactly one full wave -> EXEC all 1s
    int tile = blockIdx.x;
    int m    = lane & 15;          // matrix row handled by this lane
    int half = lane >> 4;          // which K-pair of the 16x4 A-fragment
    int b    = tile * 16 + m;

    const float* urow = acc + (size_t)user[b] * DIM;
    const float* irow = acc + ((size_t)USER_COUNT + (size_t)pos[b]) * DIM;
    const float* jrow = acc + ((size_t)USER_COUNT + (size_t)neg[b]) * DIM;

    v8f cp = {0.f, 0.f, 0.f, 0.f, 0.f, 0.f, 0.f, 0.f};
    v8f cn = {0.f, 0.f, 0.f, 0.f, 0.f, 0.f, 0.f, 0.f};

#pragma unroll
    for (int kb = 0; kb < DIM; kb += 4) {
        int k = kb + 2 * half;     // lanes 0-15: K=kb,kb+1; lanes 16-31: K=kb+2,kb+3
        v2f a, bi, bj;
        a[0]  = urow[k];  a[1]  = urow[k + 1];
        bi[0] = irow[k];  bi[1] = irow[k + 1];
        bj[0] = jrow[k];  bj[1] = jrow[k + 1];
        // D = A(16x4) * B(4x16) + C, full fp32
        cp = __builtin_amdgcn_wmma_f32_16x16x4_f32(false, a, false, bi, (short)0, cp, false, false);
        cn = __builtin_amdgcn_wmma_f32_16x16x4_f32(false, a, false, bj, (short)0, cn, false, false);
    }

    // both operands carry a 1/(L+1) factor -> scale scores by 1/16
    const float s = 1.0f / (float)((NUM_LAYER + 1) * (NUM_LAYER + 1));

    // Diagonal extraction per the 16x16 f32 C/D layout:
    //   d < 8 : lane d,    vgpr d
    //   d >= 8: lane d+16, vgpr d-8
    if (lane < 8) {
        out[tile * 16 + lane]     = cp[lane] * s;
        out[B + tile * 16 + lane] = cn[lane] * s;
    } else if (lane >= 24) {
        int d = lane - 16;
        out[tile * 16 + d]     = cp[lane - 24] * s;
        out[B + tile * 16 + d] = cn[lane - 24] * s;
    }
}

extern "C" void kernel_launch(void* const* d_in, const int* in_sizes, int n_in,
                              void* d_out, int out_size, void* d_ws, size_t ws_size,
                              hipStream_t stream) {
    const int*   user = (const int*)d_in[0];
    const int*   pos  = (const int*)d_in[1];
    const int*   neg  = (const int*)d_in[2];
    const int*   srow = (const int*)d_in[3];
    const int*   scol = (const int*)d_in[4];
    const float* sval = (const float*)d_in[5];
    const float* uemb = (const float*)d_in[6];
    const float* iemb = (const float*)d_in[7];
    float*       out  = (float*)d_out;

    const int B   = in_sizes[0];   // 8192
    const int nnz = in_sizes[3];   // 5,000,000

    const size_t bufElems = (size_t)NROWS * DIM;   // 19.2M floats = 76.8 MB
    float* acc  = (float*)d_ws;
    float* buf0 = acc + bufElems;
    float* buf1 = buf0 + bufElems;

    lgcn_init<<<4096, 256, 0, stream>>>((const float4*)uemb, (const float4*)iemb,
                                        (float4*)acc, (float4*)buf0, (float4*)buf1);

    float* cur = buf0;
    float* nxt = buf1;
    const int spmm_blocks = (nnz * 32 + 255) / 256;   // wave-per-nnz, 8 waves/block
    for (int l = 0; l < NUM_LAYER; ++l) {
        lgcn_spmm<<<spmm_blocks, 256, 0, stream>>>(srow, scol, sval, cur, nxt, nnz);
        lgcn_combine<<<4096, 256, 0, stream>>>((float4*)acc, (const float4*)nxt,
                                               (float4*)cur);
        float* t = cur; cur = nxt; nxt = t;
    }

    lgcn_score<<<B / 16, 32, 0, stream>>>(acc, user, pos, neg, out, B);
}